// LightGCN_42932493091121
// MI455X (gfx1250) — compile-verified
//
#include <hip/hip_runtime.h>

// ---------------------------------------------------------------------------
// LightGCN forward on MI455X (gfx1250).
// Sparse propagation: xn[col] += d^-1/2[row]*d^-1/2[col] * xc[row], 3 layers,
// layer-mean. ~0.46 GFLOP vs ~1.9 GB of (mostly L2-resident) traffic ->
// strictly memory/atomic bound; WMMA is inapplicable (no dense tiles).
// CDNA5 paths: async global->LDS edge staging (b128) with ASYNCcnt
// double-buffering, wave32-coalesced gathers + fp32 global atomics.
// ---------------------------------------------------------------------------

#define DIM      64
#define CHUNK    256          // edges staged per block iteration
#define MAXBLK   1536         // persistent scatter blocks

// ---------------- elementwise kernels (grid-stride) ------------------------

__global__ void k_zero(float* __restrict__ p, size_t n) {
  size_t i = (size_t)blockIdx.x * blockDim.x + threadIdx.x;
  size_t stride = (size_t)gridDim.x * blockDim.x;
  for (; i < n; i += stride) p[i] = 0.0f;
}

__global__ void k_deg(const int* __restrict__ row, float* __restrict__ deg, int E) {
  int e = blockIdx.x * blockDim.x + threadIdx.x;
  if (e < E) atomicAdd(&deg[row[e]], 1.0f);
}

__global__ void k_rsqrt(float* __restrict__ d, int n) {
  int i = blockIdx.x * blockDim.x + threadIdx.x;
  if (i < n) d[i] = rsqrtf(d[i]);   // deg==0 -> +inf, matches deg**-0.5 (unused nodes)
}

// x0 = concat(user,item); acc(out) = x0; zero the second ping-pong buffer
__global__ void k_init(const float* __restrict__ user, const float* __restrict__ item,
                       float* __restrict__ x0, float* __restrict__ out,
                       float* __restrict__ zbuf, size_t nu64, size_t ntot64) {
  size_t i = (size_t)blockIdx.x * blockDim.x + threadIdx.x;
  size_t stride = (size_t)gridDim.x * blockDim.x;
  for (; i < ntot64; i += stride) {
    float v = (i < nu64) ? user[i] : item[i - nu64];
    x0[i]   = v;
    out[i]  = v;
    zbuf[i] = 0.0f;
  }
}

// out += xn; zero the buffer the next layer will scatter into
__global__ void k_acc_zero(float* __restrict__ out, const float* __restrict__ xn,
                           float* __restrict__ zbuf, size_t n) {
  size_t i = (size_t)blockIdx.x * blockDim.x + threadIdx.x;
  size_t stride = (size_t)gridDim.x * blockDim.x;
  for (; i < n; i += stride) {
    out[i]  = out[i] + xn[i];
    zbuf[i] = 0.0f;
  }
}

// final layer: out = (out + xn) * 0.25  (layer-mean over 4 embeddings)
__global__ void k_acc_final(float* __restrict__ out, const float* __restrict__ xn,
                            size_t n) {
  size_t i = (size_t)blockIdx.x * blockDim.x + threadIdx.x;
  size_t stride = (size_t)gridDim.x * blockDim.x;
  for (; i < n; i += stride) out[i] = (out[i] + xn[i]) * 0.25f;
}

// ---------------- scatter: xn[col] += dis[row]*dis[col] * xc[row] ----------
//
// Persistent blocks, 256 threads = 8 waves. Edge indices staged into LDS with
// CDNA5 async global->LDS loads, double-buffered on ASYNCcnt (async loads
// complete in order, so waiting "<= newest" retires exactly the older chunk).
//
// USE_B128 (E % 4 == 0, true for this graph): lanes 0-63 issue one
// global_load_async_to_lds_b128 covering 4 row indices each, lanes 64-127
// likewise for cols -> 1 outstanding async op per issuing wave per chunk.
// Fallback: per-lane b32 loads (2 per thread per chunk).

template <bool USE_B128>
__global__ __launch_bounds__(256)
void k_scatter(const int* __restrict__ rowp, const int* __restrict__ colp,
               const float* __restrict__ dis, const float* __restrict__ xc,
               float* __restrict__ xn, int E, int nchunks) {
  __shared__ __align__(16) int   s_row[2][CHUNK];
  __shared__ __align__(16) int   s_col[2][CHUNK];
  __shared__ __align__(16) float s_nrm[2][CHUNK];

  const int tid = threadIdx.x;
  const int g   = tid >> 6;   // edge sub-group 0..3
  const int d   = tid & 63;   // feature dim

  auto issue_chunk = [&](int ck, int b) {
    if (USE_B128) {
      if (tid < 128) {
        const int* src = (tid < 64) ? rowp : colp;
        int* dst = (tid < 64) ? s_row[b] : s_col[b];
        int lane4 = (tid & 63) << 2;          // 0,4,...,252
        int eg = ck * CHUNK + lane4;
        if (eg < E) {                          // E%4==0 -> eg<E implies eg+3<E
          unsigned lds = (unsigned)(size_t)&dst[lane4]; // low 32b = LDS offset
          unsigned off = (unsigned)eg * 4u;
          asm volatile("global_load_async_to_lds_b128 %0, %1, %2"
                       :: "v"(lds), "v"(off), "s"((const void*)src) : "memory");
        }
      }
    } else {
      int eg = ck * CHUNK + tid;
      if (eg < E) {
        unsigned lr  = (unsigned)(size_t)&s_row[b][tid];
        unsigned lc  = (unsigned)(size_t)&s_col[b][tid];
        unsigned off = (unsigned)eg * 4u;
        asm volatile("global_load_async_to_lds_b32 %0, %1, %2"
                     :: "v"(lr), "v"(off), "s"((const void*)rowp) : "memory");
        asm volatile("global_load_async_to_lds_b32 %0, %1, %2"
                     :: "v"(lc), "v"(off), "s"((const void*)colp) : "memory");
      }
    }
  };

  int ck  = blockIdx.x;
  int buf = 0;
  if (ck < nchunks) issue_chunk(ck, 0);

  while (ck < nchunks) {
    int nxt = ck + (int)gridDim.x;
    if (nxt < nchunks) {
      issue_chunk(nxt, buf ^ 1);               // prefetch next chunk
      if (USE_B128) asm volatile("s_wait_asynccnt 0x1" ::: "memory");
      else          asm volatile("s_wait_asynccnt 0x2" ::: "memory");
    } else {
      asm volatile("s_wait_asynccnt 0x0" ::: "memory");
    }
    __syncthreads();

    // one thread per edge: symmetric norm into LDS
    {
      int eg = ck * CHUNK + tid;
      float nrm = 0.0f;
      if (eg < E) {
        int r = s_row[buf][tid];
        int c = s_col[buf][tid];
        nrm = dis[r] * dis[c];
      }
      s_nrm[buf][tid] = nrm;
    }
    __syncthreads();

    // 4 edges in flight; each 64-thread (2-wave) group owns one edge:
    // per wave32 one contiguous 128B gather + 128B of global_atomic_add_f32
    #pragma unroll 4
    for (int k = 0; k < CHUNK / 4; ++k) {
      int e = (k << 2) | g;
      float w = s_nrm[buf][e];
      if (w != 0.0f) {                          // skips tail padding only
        int r = s_row[buf][e];
        int c = s_col[buf][e];
        float v = w * xc[(size_t)r * DIM + d];  // L2-resident gather
        atomicAdd(&xn[(size_t)c * DIM + d], v);
      }
    }
    __syncthreads();   // all reads of buf done before next iter's async writes

    buf ^= 1;
    ck = nxt;
  }
}

// ---------------------------------------------------------------------------

extern "C" void kernel_launch(void* const* d_in, const int* in_sizes, int n_in,
                              void* d_out, int out_size, void* d_ws, size_t ws_size,
                              hipStream_t stream) {
  const float* user = (const float*)d_in[0];
  const float* item = (const float*)d_in[1];
  const int*   edge = (const int*)d_in[2];

  const size_t nu64  = (size_t)in_sizes[0];             // NUM_USER * 64
  const size_t ni64  = (size_t)in_sizes[1];             // NUM_ITEM * 64
  const int    N     = (int)((nu64 + ni64) / DIM);      // 150000
  const int    E     = in_sizes[2] / 2;                 // 1,200,000
  const size_t nfeat = nu64 + ni64;                     // N * 64

  const int* rowp = edge;
  const int* colp = edge + E;
  float* out = (float*)d_out;

  // workspace: [ deg/dis : Npad ][ x_a : N*64 ][ x_b : N*64 ]  (~77.4 MB)
  float* wsf  = (float*)d_ws;
  size_t Npad = ((size_t)N + 255) & ~(size_t)255;
  float* deg  = wsf;
  float* xa   = wsf + Npad;
  float* xb   = xa + nfeat;

  const int T   = 256;
  const int gN  = (N + T - 1) / T;
  const int gE  = (E + T - 1) / T;
  const int gEl = 2048;                                 // grid-stride elementwise

  // 1) degrees -> deg^-1/2 (in place)
  k_zero <<<gN, T, 0, stream>>>(deg, (size_t)N);
  k_deg  <<<gE, T, 0, stream>>>(rowp, deg, E);
  k_rsqrt<<<gN, T, 0, stream>>>(deg, N);

  // 2) x_a = concat(user,item); acc = x_a; x_b = 0
  k_init <<<gEl, T, 0, stream>>>(user, item, xa, out, xb, nu64, nfeat);

  // 3) three propagation layers, ping-pong xa/xb; zero fused into accumulate
  const int  nchunks = (E + CHUNK - 1) / CHUNK;
  const int  nblk    = nchunks < MAXBLK ? nchunks : MAXBLK;
  const bool b128    = (E & 3) == 0;
  float* xc = xa;
  float* xn = xb;
  for (int l = 0; l < 3; ++l) {
    if (b128)
      k_scatter<true> <<<nblk, T, 0, stream>>>(rowp, colp, deg, xc, xn, E, nchunks);
    else
      k_scatter<false><<<nblk, T, 0, stream>>>(rowp, colp, deg, xc, xn, E, nchunks);
    if (l < 2)
      k_acc_zero <<<gEl, T, 0, stream>>>(out, xn, xc, nfeat);  // xc reused next layer
    else
      k_acc_final<<<gEl, T, 0, stream>>>(out, xn, nfeat);
    float* t = xc; xc = xn; xn = t;
  }
}